// MSSSIMLoss_22256520528556
// MI455X (gfx1250) — compile-verified
//
#include <hip/hip_runtime.h>
#include <math.h>

typedef __attribute__((ext_vector_type(2))) float v2f;
typedef __attribute__((ext_vector_type(8))) float v8f;

// Gaussian tap weight g[t], t in [0,10], sigma=1.5 (2*sigma^2 = 4.5), normalized by invS.
__device__ __forceinline__ float tap(int t, float invS) {
    if ((unsigned)t > 10u) return 0.0f;
    float d = (float)(t - 5);
    return expf(-d * d / 4.5f) * invS;
}

// One wave (32 threads) computes one 16x16 SSIM tile via banded-matmul WMMA:
//   Temp(26x16) = Region(26x26, zero-padded) x H(26x16)   [horizontal conv]
//   Out(16x16)  = V(16x26) x Temp(26x16)                  [vertical conv]
// K padded to 28 (7 steps of K=4), M of stage-1 padded to 32 (2 blocks of 16).
__global__ __launch_bounds__(32) void ssim_partial_kernel(
    const float* __restrict__ X, const float* __restrict__ Y,
    float* __restrict__ partial, int H)
{
    __shared__ float rgx[32][34];          // region x, rows 26..31 / cols 26..33 stay zero
    __shared__ float rgy[32][34];          // region y
    __shared__ float tmp[5][32][16];       // per-field Temp (rows 26..31 are genuinely zero)

    const int lane = threadIdx.x;
    const int half = lane >> 4;            // wave32: lanes 0-15 vs 16-31
    const int n    = lane & 15;            // N column (B/D) == M row (A) for this lane

    const int T  = H >> 4;
    const int tilesPerImg = T * T;
    const int img = blockIdx.x / tilesPerImg;
    const int tid = blockIdx.x - img * tilesPerImg;
    const int tr  = tid / T;
    const int tc  = tid - tr * T;
    const long base = (long)img * H * H;
    const int r0 = tr * 16 - 5;
    const int c0 = tc * 16 - 5;

    // ---- zero region, then load 26x26 halo (zero pad outside image, as lax.conv does)
    for (int t = lane; t < 32 * 34; t += 32) {
        (&rgx[0][0])[t] = 0.0f;
        (&rgy[0][0])[t] = 0.0f;
    }
    __syncthreads();
    for (int t = lane; t < 26 * 26; t += 32) {
        int i = t / 26, j = t - i * 26;
        int gr = r0 + i, gc = c0 + j;
        float xv = 0.0f, yv = 0.0f;
        if (gr >= 0 && gr < H && gc >= 0 && gc < H) {
            long o = base + (long)gr * H + gc;
            xv = X[o];
            yv = Y[o];
        }
        rgx[i][j] = xv;
        rgy[i][j] = yv;
    }
    __syncthreads();

    // ---- normalized Gaussian taps; per-K-step banded operands.
    // Banded matrices: H[k,c] = g[k-c] (stage-1 B), V[r,k] = g[k-r] (stage-2 A).
    // With the 16x4 A layout (lane: M = lane&15, K = k0 + 2*half + j) and the
    // mirrored 4x16 B layout (lane: N = lane&15, K = k0 + 2*half + j), both
    // reduce to the same per-lane values wk[j][k] = g[(4k + 2*half + j) - (lane&15)].
    float S = 0.0f;
#pragma unroll
    for (int t = 0; t < 11; ++t) { float d = (float)(t - 5); S += expf(-d * d / 4.5f); }
    const float invS = 1.0f / S;

    float wk0[7], wk1[7];
#pragma unroll
    for (int k = 0; k < 7; ++k) {
        int ks = 4 * k + 2 * half;
        wk0[k] = tap(ks     - n, invS);
        wk1[k] = tap(ks + 1 - n, invS);
    }

    const v8f z = {0.f, 0.f, 0.f, 0.f, 0.f, 0.f, 0.f, 0.f};

    // ---- stage 1: horizontal conv of 5 fields (x, y, x*x, y*y, x*y), 2 M-blocks x 7 K-steps
    v8f s1[2][5];
#pragma unroll
    for (int m = 0; m < 2; ++m)
#pragma unroll
        for (int f = 0; f < 5; ++f) s1[m][f] = z;

#pragma unroll
    for (int m = 0; m < 2; ++m) {
        const int row = m * 16 + n;        // A-matrix row for this lane
#pragma unroll
        for (int k = 0; k < 7; ++k) {
            const int col = 4 * k + 2 * half;
            float x0 = rgx[row][col], x1 = rgx[row][col + 1];
            float y0 = rgy[row][col], y1 = rgy[row][col + 1];
            v2f b;  b[0] = wk0[k];  b[1] = wk1[k];
            v2f ax; ax[0] = x0;      ax[1] = x1;
            v2f ay; ay[0] = y0;      ay[1] = y1;
            v2f aq; aq[0] = x0 * x0; aq[1] = x1 * x1;
            v2f ar; ar[0] = y0 * y0; ar[1] = y1 * y1;
            v2f ap; ap[0] = x0 * y0; ap[1] = x1 * y1;
            s1[m][0] = __builtin_amdgcn_wmma_f32_16x16x4_f32(false, ax, false, b, (short)0, s1[m][0], false, false);
            s1[m][1] = __builtin_amdgcn_wmma_f32_16x16x4_f32(false, ay, false, b, (short)0, s1[m][1], false, false);
            s1[m][2] = __builtin_amdgcn_wmma_f32_16x16x4_f32(false, aq, false, b, (short)0, s1[m][2], false, false);
            s1[m][3] = __builtin_amdgcn_wmma_f32_16x16x4_f32(false, ar, false, b, (short)0, s1[m][3], false, false);
            s1[m][4] = __builtin_amdgcn_wmma_f32_16x16x4_f32(false, ap, false, b, (short)0, s1[m][4], false, false);
        }
    }

    // ---- scatter Temp (C/D layout: VGPR v <-> row v + 8*half per M-block) to LDS
#pragma unroll
    for (int m = 0; m < 2; ++m)
#pragma unroll
        for (int v = 0; v < 8; ++v) {
            int rowi = m * 16 + v + 8 * half;
            tmp[0][rowi][n] = s1[m][0][v];
            tmp[1][rowi][n] = s1[m][1][v];
            tmp[2][rowi][n] = s1[m][2][v];
            tmp[3][rowi][n] = s1[m][3][v];
            tmp[4][rowi][n] = s1[m][4][v];
        }
    __syncthreads();

    // ---- stage 2: vertical conv; A = V band (same wk values), B = Temp columns
    v8f outF[5];
#pragma unroll
    for (int f = 0; f < 5; ++f) {
        v8f acc = z;
#pragma unroll
        for (int k = 0; k < 7; ++k) {
            int ks = 4 * k + 2 * half;
            v2f a; a[0] = wk0[k];          a[1] = wk1[k];
            v2f b; b[0] = tmp[f][ks][n];   b[1] = tmp[f][ks + 1][n];
            acc = __builtin_amdgcn_wmma_f32_16x16x4_f32(false, a, false, b, (short)0, acc, false, false);
        }
        outF[f] = acc;
    }

    // ---- pointwise SSIM over the 256 tile pixels held across lanes/VGPRs
    const float C1 = 4.0e-4f;    // (0.01*2)^2
    const float C2 = 3.6e-3f;    // (0.03*2)^2
    float s = 0.0f;
#pragma unroll
    for (int v = 0; v < 8; ++v) {
        float m1 = outF[0][v], m2 = outF[1][v];
        float xx = outF[2][v], yy = outF[3][v], xy = outF[4][v];
        float m1s = m1 * m1, m2s = m2 * m2, m12 = m1 * m2;
        float sg1 = xx - m1s, sg2 = yy - m2s, sg12 = xy - m12;
        float num = (2.0f * m12 + C1) * (2.0f * sg12 + C2);
        float den = (m1s + m2s + C1) * (sg1 + sg2 + C2);
        s += num / (den + 1e-8f);
    }
    // deterministic wave32 butterfly reduce
#pragma unroll
    for (int off = 16; off > 0; off >>= 1) s += __shfl_xor(s, off, 32);
    if (lane == 0) partial[blockIdx.x] = s;
}

__global__ void avgpool_kernel(const float* __restrict__ src, float* __restrict__ dst,
                               int Ho, int total)
{
    int idx = blockIdx.x * blockDim.x + threadIdx.x;
    if (idx >= total) return;
    int hw  = Ho * Ho;
    int img = idx / hw;
    int rem = idx - img * hw;
    int oi  = rem / Ho;
    int oj  = rem - oi * Ho;
    int Hi  = Ho * 2;
    const float* s = src + (long)img * Hi * Hi;
    long r = 2 * oi, c = 2 * oj;
    dst[idx] = 0.25f * (s[r * Hi + c] + s[r * Hi + c + 1] +
                        s[(r + 1) * Hi + c] + s[(r + 1) * Hi + c + 1]);
}

// Fixed-order single-block reduction -> deterministic across replays.
__global__ void reduce_kernel(const float* __restrict__ in, int n, float* __restrict__ out)
{
    __shared__ float sm[256];
    float s = 0.0f;
    for (int i = threadIdx.x; i < n; i += 256) s += in[i];
    sm[threadIdx.x] = s;
    __syncthreads();
    for (int off = 128; off > 0; off >>= 1) {
        if ((int)threadIdx.x < off) sm[threadIdx.x] += sm[threadIdx.x + off];
        __syncthreads();
    }
    if (threadIdx.x == 0) *out = sm[0];
}

__global__ void final_kernel(const float* __restrict__ sums, float* __restrict__ out)
{
    const float W[5] = {0.0448f, 0.2856f, 0.3001f, 0.2363f, 0.1333f};
    float wsum = 0.0f;
    for (int i = 0; i < 5; ++i) wsum += W[i];
    float acc = 0.0f;
    int H = 512;
    for (int i = 0; i < 5; ++i) {
        float cnt = 48.0f * (float)H * (float)H;   // 16*3*H*H elements in ssim_map
        acc += (W[i] / wsum) * (sums[i] / cnt);
        H >>= 1;
    }
    out[0] = 1.0f - acc;
}

extern "C" void kernel_launch(void* const* d_in, const int* in_sizes, int n_in,
                              void* d_out, int out_size, void* d_ws, size_t ws_size,
                              hipStream_t stream)
{
    (void)in_sizes; (void)n_in; (void)out_size; (void)ws_size;
    const float* pred   = (const float*)d_in[0];
    const float* target = (const float*)d_in[1];
    float* out = (float*)d_out;
    float* ws  = (float*)d_ws;

    // workspace layout (floats)
    size_t off = 0;
    float* x1 = ws + off; off += 48L * 256 * 256;
    float* y1 = ws + off; off += 48L * 256 * 256;
    float* x2 = ws + off; off += 48L * 128 * 128;
    float* y2 = ws + off; off += 48L * 128 * 128;
    float* x3 = ws + off; off += 48L * 64 * 64;
    float* y3 = ws + off; off += 48L * 64 * 64;
    float* x4 = ws + off; off += 48L * 32 * 32;
    float* y4 = ws + off; off += 48L * 32 * 32;
    float* part = ws + off; off += 49152;          // reused for every level (stream-ordered)
    float* sums = ws + off; off += 8;

    // ---- pyramid (stream order provides the dependencies)
    {
        int Ho, total;
        Ho = 256; total = 48 * Ho * Ho;
        avgpool_kernel<<<(total + 255) / 256, 256, 0, stream>>>(pred,   x1, Ho, total);
        avgpool_kernel<<<(total + 255) / 256, 256, 0, stream>>>(target, y1, Ho, total);
        Ho = 128; total = 48 * Ho * Ho;
        avgpool_kernel<<<(total + 255) / 256, 256, 0, stream>>>(x1, x2, Ho, total);
        avgpool_kernel<<<(total + 255) / 256, 256, 0, stream>>>(y1, y2, Ho, total);
        Ho = 64; total = 48 * Ho * Ho;
        avgpool_kernel<<<(total + 255) / 256, 256, 0, stream>>>(x2, x3, Ho, total);
        avgpool_kernel<<<(total + 255) / 256, 256, 0, stream>>>(y2, y3, Ho, total);
        Ho = 32; total = 48 * Ho * Ho;
        avgpool_kernel<<<(total + 255) / 256, 256, 0, stream>>>(x3, x4, Ho, total);
        avgpool_kernel<<<(total + 255) / 256, 256, 0, stream>>>(y3, y4, Ho, total);
    }

    // ---- per-level SSIM (one wave per 16x16 tile) + deterministic reductions
    const float* xs[5] = {pred,   x1, x2, x3, x4};
    const float* ys[5] = {target, y1, y2, y3, y4};
    int H = 512;
    for (int l = 0; l < 5; ++l) {
        int T = H / 16;
        int blocks = 48 * T * T;
        ssim_partial_kernel<<<blocks, 32, 0, stream>>>(xs[l], ys[l], part, H);
        reduce_kernel<<<1, 256, 0, stream>>>(part, blocks, sums + l);
        H >>= 1;
    }
    final_kernel<<<1, 1, 0, stream>>>(sums, out);
}